// S4Model_53291954209258
// MI455X (gfx1250) — compile-verified
//
#include <hip/hip_runtime.h>
#include <hip/hip_bf16.h>

typedef __attribute__((ext_vector_type(2))) float v2f;
typedef __attribute__((ext_vector_type(8))) float v8f;

#define L_SEQ   2048
#define BATCH   16
#define DIN     64
#define H_DIM   512
#define N_ST    64
#define NLAYERS 4
#define T_CH    64
#define NCHUNK  32                  // L_SEQ / T_CH
#define LB      32768               // L_SEQ * BATCH

__device__ __forceinline__ v8f wmma4(v2f a, v2f b, v8f c) {
  // D = A(16x4,f32) x B(4x16,f32) + C(16x16,f32)
  return __builtin_amdgcn_wmma_f32_16x16x4_f32(false, a, false, b, (short)0, c,
                                               false, false);
}

// ---------------------------------------------------------------------------
// Encoder: u[h][l*16+b] = sum_k enc_W[k][h] * x[(l*16+b)][k] + enc_b[h]
// Computed as W^T @ x^T so stores into channel-major layout are coalesced.
// One wave per 16x16 output tile, K-loop in steps of 4 via WMMA f32 16x16x4.
// ---------------------------------------------------------------------------
__global__ void enc_kernel(const float* __restrict__ x,
                           const float* __restrict__ W,
                           const float* __restrict__ bvec,
                           float* __restrict__ u) {
  int wave = blockIdx.x * (blockDim.x >> 5) + (threadIdx.x >> 5);
  int lane = threadIdx.x & 31;
  int ln = lane & 15, hh = lane >> 4;
  int ntile = wave & (LB / 16 - 1);     // 2048 column tiles (l,b)
  int mtile = wave / (LB / 16);         // 32 row tiles (h)
  int row = mtile * 16 + ln;            // A-matrix row (h)
  int col = ntile * 16 + ln;            // B-matrix col (l*16+b)
  v8f acc = {};
  for (int kb = 0; kb < DIN; kb += 4) {
    int k = kb + 2 * hh;
    v2f a; a.x = W[(size_t)k * H_DIM + row];
           a.y = W[(size_t)(k + 1) * H_DIM + row];
    v2f b = *(const v2f*)(x + (size_t)col * DIN + k);
    acc = wmma4(a, b, acc);
  }
#pragma unroll
  for (int v = 0; v < 8; ++v) {
    int r = mtile * 16 + v + 8 * hh;
    u[(size_t)r * LB + ntile * 16 + ln] = acc[v] + bvec[r];
  }
}

// ---------------------------------------------------------------------------
// Per-layer precompute of M (2*C*dA^{t+1}, imag pre-negated), W (dA^{63-k}*dB)
// and dA^64, all per (h,n). One block of 64 threads per h.
// ---------------------------------------------------------------------------
__global__ void precompute_MW(const float* __restrict__ log_dt,
                              const float* __restrict__ log_A_re,
                              const float* __restrict__ A_im,
                              const float* __restrict__ C_re,
                              const float* __restrict__ C_im,
                              float* __restrict__ Mre, float* __restrict__ Mim,
                              float* __restrict__ Wre, float* __restrict__ Wim,
                              float* __restrict__ dATre, float* __restrict__ dATim) {
  int h = blockIdx.x;
  int n = threadIdx.x;
  float dt = __expf(log_dt[h]);
  float Ar = -__expf(log_A_re[h * N_ST + n]);
  float Ai = A_im[h * N_ST + n];
  float er = __expf(dt * Ar);
  float si, co; __sincosf(dt * Ai, &si, &co);
  float dAr = er * co, dAi = er * si;            // dA = exp(dt*A)
  float inv = 1.0f / (Ar * Ar + Ai * Ai);
  float nr = dAr - 1.0f, ni = dAi;               // dB = (dA-1)/A
  float dBr = (nr * Ar + ni * Ai) * inv;
  float dBi = (ni * Ar - nr * Ai) * inv;
  float Cr = C_re[h * N_ST + n], Ci = C_im[h * N_ST + n];
  float Pr = 1.f, Pi = 0.f;                      // dA^0
  size_t base = (size_t)h * (T_CH * N_ST);
  for (int j = 0; j < T_CH; ++j) {
    // W[n][63-j] = dA^j * dB
    Wre[base + n * T_CH + (63 - j)] = Pr * dBr - Pi * dBi;
    Wim[base + n * T_CH + (63 - j)] = Pr * dBi + Pi * dBr;
    float tr = Pr * dAr - Pi * dAi;
    float ti = Pr * dAi + Pi * dAr;
    Pr = tr; Pi = ti;                            // dA^{j+1}
    // M[j][n] = 2*C*dA^{j+1}; imag stored negated
    Mre[base + j * N_ST + n] =  2.0f * (Cr * Pr - Ci * Pi);
    Mim[base + j * N_ST + n] = -2.0f * (Cr * Pi + Ci * Pr);
  }
  dATre[h * N_ST + n] = Pr;                      // dA^64
  dATim[h * N_ST + n] = Pi;
}

// ---------------------------------------------------------------------------
// Convolution taps: Ktap[h][j] = 2*sum_n Re(C*dB*dA^j)  (+ Dskip at j==0).
// One wave per (h,j), 2 n-values per lane, shuffle reduction.
// ---------------------------------------------------------------------------
__global__ void precompute_K(const float* __restrict__ log_dt,
                             const float* __restrict__ log_A_re,
                             const float* __restrict__ A_im,
                             const float* __restrict__ C_re,
                             const float* __restrict__ C_im,
                             const float* __restrict__ Dskip,
                             float* __restrict__ Ktap) {
  int wid = blockIdx.x * (blockDim.x >> 5) + (threadIdx.x >> 5);
  int lane = threadIdx.x & 31;
  int h = wid >> 6;
  int j = wid & 63;
  float dt = __expf(log_dt[h]);
  float acc = 0.f;
#pragma unroll
  for (int t = 0; t < 2; ++t) {
    int n = lane + 32 * t;
    float Ar = -__expf(log_A_re[h * N_ST + n]);
    float Ai = A_im[h * N_ST + n];
    float er = __expf(dt * Ar);
    float si, co; __sincosf(dt * Ai, &si, &co);
    float dAr = er * co, dAi = er * si;
    float inv = 1.0f / (Ar * Ar + Ai * Ai);
    float nr = dAr - 1.f, ni = dAi;
    float dBr = (nr * Ar + ni * Ai) * inv;
    float dBi = (ni * Ar - nr * Ai) * inv;
    float Cr = C_re[h * N_ST + n], Ci = C_im[h * N_ST + n];
    float Gr = Cr * dBr - Ci * dBi, Gi = Cr * dBi + Ci * dBr;   // C*dB
    float pj = (float)j;                                        // dA^j closed form
    float erj = __expf(pj * dt * Ar);
    float sj, cj; __sincosf(pj * dt * Ai, &sj, &cj);
    acc += Gr * (erj * cj) - Gi * (erj * sj);
  }
  for (int m = 16; m >= 1; m >>= 1) acc += __shfl_xor(acc, m, 32);
  if (lane == 0) Ktap[h * 64 + j] = 2.f * acc + (j == 0 ? Dskip[h] : 0.f);
}

// ---------------------------------------------------------------------------
// Chunk state gather: cs[h][c][n][b] = sum_k W[h][n][k] * u[h][c*64+k][b]
// (complex W, real u). One block per (h,c); 4 waves = 4 row-tiles of N.
// ---------------------------------------------------------------------------
__global__ void chunksum_kernel(const float* __restrict__ u,
                                const float* __restrict__ Wre,
                                const float* __restrict__ Wim,
                                float* __restrict__ csre,
                                float* __restrict__ csim) {
  int hc = blockIdx.x;               // h*32 + c
  int h = hc >> 5, c = hc & 31;
  int wv = threadIdx.x >> 5;         // mtile 0..3
  int lane = threadIdx.x & 31;
  int ln = lane & 15, hh = lane >> 4;
  const float* Ub = u + (size_t)h * LB + c * (T_CH * BATCH);
  const float* Ar = Wre + (size_t)h * 4096 + (wv * 16 + ln) * 64;
  const float* Aw = Wim + (size_t)h * 4096 + (wv * 16 + ln) * 64;
  v8f ar = {}, ai = {};
  for (int kb = 0; kb < T_CH; kb += 4) {
    int k = kb + 2 * hh;
    v2f b; b.x = Ub[k * 16 + ln]; b.y = Ub[(k + 1) * 16 + ln];
    v2f a1 = *(const v2f*)(Ar + k);
    v2f a2 = *(const v2f*)(Aw + k);
    ar = wmma4(a1, b, ar);
    ai = wmma4(a2, b, ai);
  }
  size_t ob = (size_t)hc * (N_ST * BATCH);
#pragma unroll
  for (int v = 0; v < 8; ++v) {
    int nrow = wv * 16 + v + 8 * hh;
    csre[ob + nrow * 16 + ln] = ar[v];
    csim[ob + nrow * 16 + ln] = ai[v];
  }
}

// ---------------------------------------------------------------------------
// Cross-chunk scan (32 serial steps, elementwise complex). Rewrites cs[c]
// in place with the state ENTERING chunk c; emits final state to d_out.
// ---------------------------------------------------------------------------
__global__ void scan_kernel(const float* __restrict__ dATre,
                            const float* __restrict__ dATim,
                            float* __restrict__ csre, float* __restrict__ csim,
                            float* __restrict__ outSre, float* __restrict__ outSim) {
  int tid = blockIdx.x * blockDim.x + threadIdx.x;   // B*H*N = 524288
  int b = tid & 15;
  int n = (tid >> 4) & 63;
  int h = tid >> 10;
  float ar = dATre[h * N_ST + n], ai = dATim[h * N_ST + n];   // dA^64
  float sr = 0.f, si = 0.f;
  for (int c = 0; c < NCHUNK; ++c) {
    size_t idx = ((size_t)(h * 32 + c) * 64 + n) * 16 + b;
    float xr = csre[idx], xi = csim[idx];
    csre[idx] = sr; csim[idx] = si;                  // S0 entering chunk c
    float tr = ar * sr - ai * si + xr;
    float ti = ar * si + ai * sr + xi;
    sr = tr; si = ti;
  }
  size_t so = ((size_t)b * H_DIM + h) * N_ST + n;    // states[layer][b][h][n]
  outSre[so] = sr; outSim[so] = si;
}

// ---------------------------------------------------------------------------
// Intra-chunk output (in place over u):
//   Y = Tril-Toeplitz(Ktap) @ U  +  Mre @ S0re  +  (-Mim) @ S0im
// U tile staged in LDS so the in-place write is race-free.
// ---------------------------------------------------------------------------
__global__ void yout_kernel(const float* __restrict__ Ktap,
                            const float* __restrict__ Mre,
                            const float* __restrict__ Mim,
                            const float* __restrict__ csre,
                            const float* __restrict__ csim,
                            float* __restrict__ u) {
  __shared__ float Uld[T_CH * BATCH];
  int hc = blockIdx.x;
  int h = hc >> 5, c = hc & 31;
  int wv = threadIdx.x >> 5, lane = threadIdx.x & 31;
  int ln = lane & 15, hh = lane >> 4;
  float* Ug = u + (size_t)h * LB + c * (T_CH * BATCH);
  for (int i = threadIdx.x; i < T_CH * BATCH; i += 128) Uld[i] = Ug[i];
  __syncthreads();
  int ti = wv * 16 + ln;                                 // output row (time)
  const float* Kt = Ktap + h * 64;
  const float* Mr = Mre + (size_t)h * 4096 + ti * 64;
  const float* Mi = Mim + (size_t)h * 4096 + ti * 64;
  const float* Sr = csre + (size_t)hc * (N_ST * BATCH);
  const float* Si = csim + (size_t)hc * (N_ST * BATCH);
  v8f acc = {};
  for (int kb = 0; kb < 64; kb += 4) {
    int k = kb + 2 * hh;
    v2f bu;  bu.x  = Uld[k * 16 + ln]; bu.y  = Uld[(k + 1) * 16 + ln];
    v2f bsr; bsr.x = Sr[k * 16 + ln];  bsr.y = Sr[(k + 1) * 16 + ln];
    v2f bsi; bsi.x = Si[k * 16 + ln];  bsi.y = Si[(k + 1) * 16 + ln];
    v2f at;  at.x = (k     <= ti) ? Kt[ti - k]     : 0.f;
             at.y = (k + 1 <= ti) ? Kt[ti - k - 1] : 0.f;
    v2f am = *(const v2f*)(Mr + k);
    v2f ami = *(const v2f*)(Mi + k);
    acc = wmma4(at, bu, acc);
    acc = wmma4(am, bsr, acc);
    acc = wmma4(ami, bsi, acc);
  }
#pragma unroll
  for (int v = 0; v < 8; ++v) {
    int row = wv * 16 + v + 8 * hh;
    Ug[row * 16 + ln] = acc[v];
  }
}

// ---------------------------------------------------------------------------
// Final transpose: channel-major u[H][L*B] -> d_out[L*B][H], LDS-tiled.
// ---------------------------------------------------------------------------
__global__ void transpose_kernel(const float* __restrict__ uT,
                                 float* __restrict__ out) {
  __shared__ float tile[64][65];
  int hb = blockIdx.x & 7;            // H/64
  int cb = blockIdx.x >> 3;           // LB/64
  int tx = threadIdx.x & 63;
  int ty = threadIdx.x >> 6;          // 0..3
  int h0 = hb * 64, c0 = cb * 64;
#pragma unroll
  for (int i = 0; i < 16; ++i) {
    int r = ty + i * 4;
    tile[r][tx] = uT[(size_t)(h0 + r) * LB + c0 + tx];
  }
  __syncthreads();
#pragma unroll
  for (int i = 0; i < 16; ++i) {
    int r = ty + i * 4;
    out[(size_t)(c0 + r) * H_DIM + h0 + tx] = tile[tx][r];
  }
}

extern "C" void kernel_launch(void* const* d_in, const int* in_sizes, int n_in,
                              void* d_out, int out_size, void* d_ws, size_t ws_size,
                              hipStream_t stream) {
  const float* x        = (const float*)d_in[0];
  const float* enc_W    = (const float*)d_in[1];
  const float* enc_b    = (const float*)d_in[2];
  const float* log_dt   = (const float*)d_in[3];
  const float* log_A_re = (const float*)d_in[4];
  const float* A_im     = (const float*)d_in[5];
  const float* C_re     = (const float*)d_in[6];
  const float* C_im     = (const float*)d_in[7];
  const float* Dskip    = (const float*)d_in[8];
  float* out = (float*)d_out;
  float* ws  = (float*)d_ws;

  // workspace layout (floats)
  float* u     = ws;                    // 16,777,216 : activations [H][L][B]
  float* Mre   = u + 16777216;          //  2,097,152 : [H][64][64]
  float* Mim   = Mre + 2097152;
  float* Wre   = Mim + 2097152;
  float* Wim   = Wre + 2097152;
  float* Ktap  = Wim + 2097152;         //     32,768 : [H][64]
  float* dATre = Ktap + 32768;
  float* dATim = dATre + 32768;
  float* csre  = dATim + 32768;         // 16,777,216 : [H][32][64][16]
  float* csim  = csre + 16777216;       // 16,777,216

  float* statesRe = out + 16777216;                 // [4][B][H][N]
  float* statesIm = statesRe + NLAYERS * 524288;    // [4][B][H][N]

  // encoder: 65536 tile-waves, 4 waves/block
  enc_kernel<<<16384, 128, 0, stream>>>(x, enc_W, enc_b, u);

  for (int layer = 0; layer < NLAYERS; ++layer) {
    const float* ldt = log_dt   + layer * H_DIM;
    const float* lAr = log_A_re + layer * H_DIM * N_ST;
    const float* lAi = A_im     + layer * H_DIM * N_ST;
    const float* lCr = C_re     + layer * H_DIM * N_ST;
    const float* lCi = C_im     + layer * H_DIM * N_ST;
    const float* lD  = Dskip    + layer * H_DIM;

    precompute_MW<<<H_DIM, 64, 0, stream>>>(ldt, lAr, lAi, lCr, lCi,
                                            Mre, Mim, Wre, Wim, dATre, dATim);
    precompute_K<<<8192, 128, 0, stream>>>(ldt, lAr, lAi, lCr, lCi, lD, Ktap);
    chunksum_kernel<<<H_DIM * NCHUNK, 128, 0, stream>>>(u, Wre, Wim, csre, csim);
    scan_kernel<<<2048, 256, 0, stream>>>(dATre, dATim, csre, csim,
                                          statesRe + layer * 524288,
                                          statesIm + layer * 524288);
    yout_kernel<<<H_DIM * NCHUNK, 128, 0, stream>>>(Ktap, Mre, Mim, csre, csim, u);
  }

  transpose_kernel<<<4096, 256, 0, stream>>>(u, out);
}